// DRCL_54004918779968
// MI455X (gfx1250) — compile-verified
//
#include <hip/hip_runtime.h>
#include <math.h>

// ---------------- problem constants (from reference) ----------------
#define Bn   4
#define Dn   256
#define Hn   128
#define Wn   128
#define HWn  (Hn * Wn)        // 16384
#define NRk  32
#define NSk  64
#define TAUF 0.1f
#define GWf  0.5f
#define NEGF (-1e30f)

typedef __attribute__((ext_vector_type(2))) float v2f;
typedef __attribute__((ext_vector_type(8))) float v8f;
typedef __attribute__((ext_vector_type(4))) int   v4i;

typedef __attribute__((address_space(1))) v4i* gv4i_ptr;   // global
typedef __attribute__((address_space(3))) v4i* lv4i_ptr;   // LDS

// ---------------------------------------------------------------------
// CDNA5 async global->LDS copy (ASYNCcnt-tracked), 16 bytes per lane.
// Builtin signature (from compiler diagnostic): (v4i AS1*, v4i AS3*, int, int)
// ---------------------------------------------------------------------
__device__ __forceinline__ void async_copy_b128(const float* g, float* l) {
#if __has_builtin(__builtin_amdgcn_global_load_async_to_lds_b128)
  __builtin_amdgcn_global_load_async_to_lds_b128(
      (gv4i_ptr)(v4i*)(void*)g,
      (lv4i_ptr)(v4i*)(void*)l,
      0, 0);
#else
  const unsigned ldsoff = (unsigned)(unsigned long long)l;  // flat addr low 32 = LDS offset
  asm volatile("global_load_async_to_lds_b128 %0, %1, off"
               :: "v"(ldsoff), "v"(g) : "memory");
#endif
}

__device__ __forceinline__ void wait_async_le2() {
#if __has_builtin(__builtin_amdgcn_s_wait_asynccnt)
  __builtin_amdgcn_s_wait_asynccnt(2);
#else
  asm volatile("s_wait_asynccnt 0x2" ::: "memory");
#endif
}

__device__ __forceinline__ void wait_async_0() {
#if __has_builtin(__builtin_amdgcn_s_wait_asynccnt)
  __builtin_amdgcn_s_wait_asynccnt(0);
#else
  asm volatile("s_wait_asynccnt 0x0" ::: "memory");
#endif
}

// =====================================================================
// Kernel 1/3: fp32 WMMA GEMM  C[b] = A[E,K] @ B[b][K,HW]  (+bias, opt BN+ReLU)
//   One wave computes a 16(M) x 64(N) tile; K stepped by 4 via
//   V_WMMA_F32_16X16X4_F32. The 4x64 B tile is double-buffered in LDS with
//   async B128 copies (2 issues/stage; s_wait_asynccnt 0x2 releases the
//   current buffer while the next stage is in flight). Rows padded to 72
//   floats (288B, 16B aligned) to avoid LDS bank conflicts between the
//   kb=0/1 (lanes 0-15) and kb=2/3 (lanes 16-31) halves.
//   useBN fuses y = relu(sc[k]*x + sh[k]) into the LDS->fragment read.
// =====================================================================
__global__ __launch_bounds__(32) void gemm16_wmma(
    const float* __restrict__ A, const float* __restrict__ Bm,
    float* __restrict__ C, const float* __restrict__ bias,
    const float* __restrict__ sc, const float* __restrict__ sh, int useBN)
{
  __shared__ __align__(16) float bstage[2][4][72];

  const int lane = threadIdx.x;
  const int n0 = blockIdx.x * 64;
  const int m0 = blockIdx.y * 16;
  const int b  = blockIdx.z;
  const float* __restrict__ Bp = Bm + (size_t)b * Dn * HWn;
  float* __restrict__ Cp = C + (size_t)b * Dn * HWn;

  // async-copy mapping: 2 issues x 32 lanes x 16B cover the 4x64 f32 tile
  const int r0 = lane >> 4,        c0 = (lane & 15) << 2;         // rows 0..1
  const int r1 = (lane + 32) >> 4, c1 = c0;                        // rows 2..3

  v8f acc[4] = {};
  const int mrow = m0 + (lane & 15);
  const int kb   = (lane >> 4) << 1;      // 0 or 2
  const int ncol = lane & 15;

  // prologue: stage k=0 into buffer 0
  async_copy_b128(Bp + (size_t)r0 * HWn + n0 + c0, &bstage[0][r0][c0]);
  async_copy_b128(Bp + (size_t)r1 * HWn + n0 + c1, &bstage[0][r1][c1]);

  int buf = 0;
  for (int k = 0; k < Dn; k += 4) {
    const bool more = (k + 4) < Dn;
    if (more) {
      async_copy_b128(Bp + (size_t)(k + 4 + r0) * HWn + n0 + c0, &bstage[buf ^ 1][r0][c0]);
      async_copy_b128(Bp + (size_t)(k + 4 + r1) * HWn + n0 + c1, &bstage[buf ^ 1][r1][c1]);
    }

    v2f a;
    a.x = A[mrow * Dn + (k + kb)];
    a.y = A[mrow * Dn + (k + kb + 1)];
    float s0 = 1.f, h0 = 0.f, s1 = 1.f, h1 = 0.f;
    if (useBN) { s0 = sc[k + kb]; h0 = sh[k + kb]; s1 = sc[k + kb + 1]; h1 = sh[k + kb + 1]; }

    if (more) wait_async_le2(); else wait_async_0();

#pragma unroll
    for (int t = 0; t < 4; ++t) {
      const int tc = t * 16 + ncol;
      float bx = bstage[buf][kb][tc];
      float by = bstage[buf][kb + 1][tc];
      if (useBN) { bx = fmaxf(fmaf(s0, bx, h0), 0.f); by = fmaxf(fmaf(s1, by, h1), 0.f); }
      v2f bf; bf.x = bx; bf.y = by;
      acc[t] = __builtin_amdgcn_wmma_f32_16x16x4_f32(
          /*neg_a=*/false, a, /*neg_b=*/false, bf,
          /*c_mod=*/(short)0, acc[t], /*reuse_a=*/false, /*reuse_b=*/false);
    }
    buf ^= 1;
  }

  const int rbase = m0 + ((lane >> 4) << 3);
#pragma unroll
  for (int t = 0; t < 4; ++t) {
    const int n = n0 + t * 16 + ncol;
#pragma unroll
    for (int i = 0; i < 8; ++i) {
      const int row = rbase + i;
      Cp[(size_t)row * HWn + n] = acc[t][i] + bias[row];
    }
  }
}

// =====================================================================
// Kernel 2: BN batch stats per channel -> folded scale/shift
// =====================================================================
__global__ __launch_bounds__(256) void bn_stats(
    const float* __restrict__ x, const float* __restrict__ gamma,
    const float* __restrict__ beta, float* __restrict__ sc, float* __restrict__ sh)
{
  const int e = blockIdx.x;
  const int tid = threadIdx.x;
  float s = 0.f, ss = 0.f;
  for (int b = 0; b < Bn; ++b) {
    const float* p = x + ((size_t)b * Dn + e) * HWn;
    for (int i = tid; i < HWn; i += 256) { float v = p[i]; s += v; ss += v * v; }
  }
  __shared__ float r1[256], r2[256];
  r1[tid] = s; r2[tid] = ss; __syncthreads();
  for (int st = 128; st > 0; st >>= 1) {
    if (tid < st) { r1[tid] += r1[tid + st]; r2[tid] += r2[tid + st]; }
    __syncthreads();
  }
  if (tid == 0) {
    const float inv_n = 1.f / (float)(Bn * HWn);
    float mu = r1[0] * inv_n;
    float var = r2[0] * inv_n - mu * mu;
    float scv = gamma[e] / sqrtf(var + 1e-5f);
    sc[e] = scv;
    sh[e] = beta[e] - mu * scv;
  }
}

// =====================================================================
// Kernel 4: valid-pixel masks + per-batch class counts
// =====================================================================
__global__ __launch_bounds__(256) void masks_kernel(
    const int* __restrict__ labels, const float* __restrict__ po,
    const float* __restrict__ pa, const float* __restrict__ unc,
    int* __restrict__ m1, int* __restrict__ m0, int* __restrict__ cnt)
{
  const int idx = blockIdx.x * 256 + threadIdx.x;
  if (idx >= Bn * HWn) return;
  const int b = idx / HWn, p = idx - b * HWn;
  const size_t base = (size_t)b * 2 * HWn + p;
  const int ao = (po[base + HWn] > po[base]) ? 1 : 0;
  const int aa = (pa[base + HWn] > pa[base]) ? 1 : 0;
  const bool rel = (ao == aa);
  const bool diff = unc[idx] > 0.5f;
  const bool valid = rel && diff;
  const int lab = labels[idx];
  const int v1 = (valid && lab == 1) ? 1 : 0;
  const int v0 = (valid && lab == 0) ? 1 : 0;
  m1[idx] = v1; m0[idx] = v0;
  if (v1) atomicAdd(&cnt[b * 2 + 0], 1);
  if (v0) atomicAdd(&cnt[b * 2 + 1], 1);
}

__global__ void init_cnt(int* cnt) { if (threadIdx.x < 8) cnt[threadIdx.x] = 0; }

// =====================================================================
// Kernel 5: exact iterative top-k selections per (b, class)
// =====================================================================
__device__ __forceinline__ void block_topk(
    float* loc, float* rv, int* ri, int* sgI, float* sgV,
    int K, int* outIdx, float* outValid, int tid)
{
  for (int it = 0; it < K; ++it) {
    float bv = loc[0]; int bj = 0;
    for (int j = 1; j < 64; ++j) if (loc[j] > bv) { bv = loc[j]; bj = j; }
    rv[tid] = bv; ri[tid] = tid * 64 + bj;
    __syncthreads();
    for (int s = 128; s > 0; s >>= 1) {
      if (tid < s && rv[tid + s] > rv[tid]) { rv[tid] = rv[tid + s]; ri[tid] = ri[tid + s]; }
      __syncthreads();
    }
    if (tid == 0) { *sgI = ri[0]; *sgV = rv[0]; }
    __syncthreads();
    const int gi = *sgI; const float gv = *sgV;
    if (tid == 0) { outIdx[it] = gi; outValid[it] = (gv > -1e29f) ? 1.f : 0.f; }
    if (tid == (gi >> 6)) loc[gi & 63] = NEGF;
    __syncthreads();
  }
}

__global__ __launch_bounds__(256) void select_k(
    const int* __restrict__ m1, const int* __restrict__ m0,
    const float* __restrict__ r_anc, const float* __restrict__ r_pos,
    const float* __restrict__ r_neg, const float* __restrict__ unc,
    int* __restrict__ selI, float* __restrict__ selV)
{
  const int bc = blockIdx.x;
  const int b = bc >> 1, c = bc & 1;
  const int tid = threadIdx.x;
  const int* am = (c == 0 ? m1 : m0) + b * HWn;
  const int* nm = (c == 0 ? m0 : m1) + b * HWn;
  const float* ra = r_anc + (size_t)bc * HWn;
  const float* rp = r_pos + (size_t)bc * HWn;
  const float* rn = r_neg + (size_t)bc * HWn;
  int* oI = selI + bc * 160;
  float* oV = selV + bc * 160;

  __shared__ float rv[256]; __shared__ int ri[256];
  __shared__ int gI; __shared__ float gV;
  __shared__ int cI[128]; __shared__ float cV[128];
  float loc[64];

  // ---- anchors: top-NR of masked r_anc ----
  for (int j = 0; j < 64; ++j) { const int g = tid * 64 + j; loc[j] = am[g] ? ra[g] : NEGF; }
  __syncthreads();
  block_topk(loc, rv, ri, &gI, &gV, NRk, oI, oV, tid);

  // ---- positives: top-2NS of r_pos, then top-NS by uncertainty ----
  for (int j = 0; j < 64; ++j) { const int g = tid * 64 + j; loc[j] = am[g] ? rp[g] : NEGF; }
  __syncthreads();
  block_topk(loc, rv, ri, &gI, &gV, 2 * NSk, cI, cV, tid);
  __syncthreads();
  if (tid < 128) rv[tid] = (cV[tid] > 0.f) ? unc[(size_t)b * HWn + cI[tid]] : NEGF;
  __syncthreads();
  if (tid == 0) {
    for (int it = 0; it < NSk; ++it) {
      float bv = rv[0]; int bj = 0;
      for (int j = 1; j < 128; ++j) if (rv[j] > bv) { bv = rv[j]; bj = j; }
      oI[32 + it] = cI[bj]; oV[32 + it] = cV[bj];
      rv[bj] = NEGF - 1e29f;
    }
  }
  __syncthreads();

  // ---- negatives: same with nm / r_neg ----
  for (int j = 0; j < 64; ++j) { const int g = tid * 64 + j; loc[j] = nm[g] ? rn[g] : NEGF; }
  __syncthreads();
  block_topk(loc, rv, ri, &gI, &gV, 2 * NSk, cI, cV, tid);
  __syncthreads();
  if (tid < 128) rv[tid] = (cV[tid] > 0.f) ? unc[(size_t)b * HWn + cI[tid]] : NEGF;
  __syncthreads();
  if (tid == 0) {
    for (int it = 0; it < NSk; ++it) {
      float bv = rv[0]; int bj = 0;
      for (int j = 1; j < 128; ++j) if (rv[j] > bv) { bv = rv[j]; bj = j; }
      oI[96 + it] = cI[bj]; oV[96 + it] = cV[bj];
      rv[bj] = NEGF - 1e29f;
    }
  }
}

// =====================================================================
// Kernel 6: local InfoNCE pair loss per (b, class)
// =====================================================================
__global__ __launch_bounds__(256) void local_loss(
    const float* __restrict__ proj, const int* __restrict__ selI,
    const float* __restrict__ selV, const int* __restrict__ cnt,
    float* __restrict__ bl, float* __restrict__ incf)
{
  const int bc = blockIdx.x;
  const int b = bc >> 1, c = bc & 1;
  const int tid = threadIdx.x;
  const int* oI = selI + bc * 160;
  const float* oV = selV + bc * 160;

  __shared__ int sIdx[160]; __shared__ float sVal[160]; __shared__ float invn[160];
  __shared__ float pp[256], nn[256];
  __shared__ float perw[32], afv[32];

  if (tid < 160) { sIdx[tid] = oI[tid]; sVal[tid] = oV[tid]; }
  __syncthreads();

  const float* __restrict__ F = proj + (size_t)b * Dn * HWn;  // [D, HW]
  if (tid < 160) {
    const int idx = sIdx[tid];
    float ss = 0.f;
    for (int d = 0; d < Dn; ++d) { const float v = F[(size_t)d * HWn + idx]; ss += v * v; }
    invn[tid] = 1.f / fmaxf(sqrtf(ss), 1e-12f);
  }
  __syncthreads();

  const int r = tid >> 3;        // anchor 0..31
  const int u = tid & 7;
  const int aidx = sIdx[r];
  const float ainv = invn[r];
  float sp = 0.f, sn = 0.f;
  for (int s = u; s < NSk; s += 8) {
    const int pidx = sIdx[32 + s];
    const int nidx = sIdx[96 + s];
    float dp = 0.f, dn = 0.f;
    for (int d = 0; d < Dn; ++d) {
      const float av = F[(size_t)d * HWn + aidx];
      dp += av * F[(size_t)d * HWn + pidx];
      dn += av * F[(size_t)d * HWn + nidx];
    }
    sp += expf(dp * ainv * invn[32 + s] / TAUF) * sVal[32 + s];
    sn += expf(dn * ainv * invn[96 + s] / TAUF) * sVal[96 + s];
  }
  pp[tid] = sp; nn[tid] = sn;
  __syncthreads();

  const int amc = cnt[b * 2 + (c == 0 ? 0 : 1)];
  const int nmc = cnt[b * 2 + (c == 0 ? 1 : 0)];
  const bool inc = (amc >= 1) && (nmc >= 1);
  if (tid < 32) {
    float P = 0.f, N = 0.f;
    for (int k = 0; k < 8; ++k) { P += pp[tid * 8 + k]; N += nn[tid * 8 + k]; }
    if (!inc) P += 1.f;
    const float per = -logf(P / (P + N + 1e-8f));
    perw[tid] = per * sVal[tid];
    afv[tid] = sVal[tid];
  }
  __syncthreads();
  if (tid == 0) {
    float s1 = 0.f, s2 = 0.f;
    for (int i = 0; i < 32; ++i) { s1 += perw[i]; s2 += afv[i]; }
    bl[bc] = inc ? (s1 / fmaxf(s2, 1.f)) : 0.f;
    incf[bc] = inc ? 1.f : 0.f;
  }
}

// =====================================================================
// Kernel 7: masked channel sums for prototypes
// =====================================================================
__global__ __launch_bounds__(256) void proto_kernel(
    const float* __restrict__ proj, const int* __restrict__ m1,
    const int* __restrict__ m0, float* __restrict__ msf, float* __restrict__ msb)
{
  const int e = blockIdx.x, b = blockIdx.y;
  const int tid = threadIdx.x;
  const float* p = proj + ((size_t)b * Dn + e) * HWn;
  const int* a1 = m1 + b * HWn;
  const int* a0 = m0 + b * HWn;
  float s1 = 0.f, s0 = 0.f;
  for (int i = tid; i < HWn; i += 256) {
    const float v = p[i];
    if (a1[i]) s1 += v;
    if (a0[i]) s0 += v;
  }
  __shared__ float r1[256], r0[256];
  r1[tid] = s1; r0[tid] = s0; __syncthreads();
  for (int st = 128; st > 0; st >>= 1) {
    if (tid < st) { r1[tid] += r1[tid + st]; r0[tid] += r0[tid + st]; }
    __syncthreads();
  }
  if (tid == 0) { msf[b * Dn + e] = r1[0]; msb[b * Dn + e] = r0[0]; }
}

// =====================================================================
// Kernel 8: global memory-bank loss + final combine
// =====================================================================
__global__ __launch_bounds__(256) void final_loss(
    const float* __restrict__ msf, const float* __restrict__ msb,
    const int* __restrict__ cnt, const float* __restrict__ bl,
    const float* __restrict__ incf, float* __restrict__ out)
{
  __shared__ float qf[Bn][Dn], qb[Bn][Dn];
  const int tid = threadIdx.x;
  for (int j = tid; j < Bn * Dn; j += 256) {
    const int b = j >> 8, d = j & 255;
    qf[b][d] = msf[j] / fmaxf((float)cnt[b * 2 + 0], 1.f);
    qb[b][d] = msb[j] / fmaxf((float)cnt[b * 2 + 1], 1.f);
  }
  __syncthreads();
  if (tid < 8) {
    const int b = tid & 3;
    float* v = (tid < 4) ? qf[b] : qb[b];
    float ss = 0.f;
    for (int d = 0; d < Dn; ++d) ss += v[d] * v[d];
    const float inv = 1.f / fmaxf(sqrtf(ss), 1e-12f);
    for (int d = 0; d < Dn; ++d) v[d] *= inv;
  }
  __syncthreads();
  if (tid == 0) {
    float vg[Bn];
    for (int b = 0; b < Bn; ++b)
      vg[b] = (cnt[b * 2 + 0] >= 1 && cnt[b * 2 + 1] >= 1) ? 1.f : 0.f;
    float lg_sum = 0.f, vg_sum = 0.f;
    for (int i = 0; i < Bn; ++i) {
      float nf = 0.f, nb = 0.f;
      for (int j = 0; j <= i; ++j) {
        if (vg[j] > 0.f) {
          float d1 = 0.f, d2 = 0.f;
          for (int d = 0; d < Dn; ++d) { d1 += qb[j][d] * qf[i][d]; d2 += qf[j][d] * qb[i][d]; }
          nf += expf(d1 / TAUF);
          nb += expf(d2 / TAUF);
        }
      }
      float sf = 0.f, sb = 0.f;
      for (int d = 0; d < Dn; ++d) { sf += qf[i][d] * qf[i][d]; sb += qb[i][d] * qb[i][d]; }
      const float pf = expf(sf / TAUF), pb = expf(sb / TAUF);
      const float lg = -logf(pf / (pf + nf + 1e-8f)) - logf(pb / (pb + nb + 1e-8f));
      lg_sum += vg[i] * lg;
      vg_sum += vg[i];
    }
    const float l_global = lg_sum / fmaxf(vg_sum, 1.f);
    float bls = 0.f, incs = 0.f;
    for (int i = 0; i < 2 * Bn; ++i) { bls += bl[i]; incs += incf[i]; }
    const float l_local = bls / fmaxf(incs, 1.f);
    out[0] = l_local + GWf * l_global;
    out[1] = l_local;
    out[2] = l_global;
  }
}

// =====================================================================
// Host entry
// =====================================================================
extern "C" void kernel_launch(void* const* d_in, const int* in_sizes, int n_in,
                              void* d_out, int out_size, void* d_ws, size_t ws_size,
                              hipStream_t stream) {
  (void)in_sizes; (void)n_in; (void)out_size; (void)ws_size;
  const float* feat     = (const float*)d_in[0];
  const int*   labels   = (const int*)d_in[1];
  const float* prob_ori = (const float*)d_in[2];
  const float* prob_aug = (const float*)d_in[3];
  const float* unc      = (const float*)d_in[4];
  const float* r_anc    = (const float*)d_in[5];
  const float* r_pos    = (const float*)d_in[6];
  const float* r_neg    = (const float*)d_in[7];
  const float* w1       = (const float*)d_in[8];
  const float* b1       = (const float*)d_in[9];
  const float* gamma    = (const float*)d_in[10];
  const float* beta     = (const float*)d_in[11];
  const float* w2       = (const float*)d_in[12];
  const float* b2       = (const float*)d_in[13];

  // ---- workspace layout ----
  float* x    = (float*)d_ws;                        // [Bn,Dn,HWn]
  float* proj = x + (size_t)Bn * Dn * HWn;           // [Bn,Dn,HWn]
  float* sc   = proj + (size_t)Bn * Dn * HWn;        // [Dn]
  float* sh   = sc + Dn;                             // [Dn]
  float* msf  = sh + Dn;                             // [Bn,Dn]
  float* msb  = msf + Bn * Dn;                       // [Bn,Dn]
  float* blv  = msb + Bn * Dn;                       // [8]
  float* incv = blv + 8;                             // [8]
  float* selV = incv + 8;                            // [8*160]
  int*   ip   = (int*)(selV + 8 * 160);
  int*   m1   = ip;  ip += Bn * HWn;
  int*   m0   = ip;  ip += Bn * HWn;
  int*   cnt  = ip;  ip += 8;
  int*   selI = ip;                                  // [8*160]

  init_cnt<<<1, 32, 0, stream>>>(cnt);

  // GEMM1: x = w1 @ feat + b1
  gemm16_wmma<<<dim3(HWn / 64, Dn / 16, Bn), 32, 0, stream>>>(
      w1, feat, x, b1, nullptr, nullptr, 0);

  // BN batch stats -> folded scale/shift
  bn_stats<<<Dn, 256, 0, stream>>>(x, gamma, beta, sc, sh);

  // GEMM2: proj = w2 @ relu(BN(x)) + b2  (BN+ReLU fused into LDS read)
  gemm16_wmma<<<dim3(HWn / 64, Dn / 16, Bn), 32, 0, stream>>>(
      w2, x, proj, b2, sc, sh, 1);

  // masks + counts
  masks_kernel<<<(Bn * HWn) / 256, 256, 0, stream>>>(
      labels, prob_ori, prob_aug, unc, m1, m0, cnt);

  // top-k selections per (b, class)
  select_k<<<2 * Bn, 256, 0, stream>>>(m1, m0, r_anc, r_pos, r_neg, unc, selI, selV);

  // local pair loss
  local_loss<<<2 * Bn, 256, 0, stream>>>(proj, selI, selV, cnt, blv, incv);

  // prototypes
  proto_kernel<<<dim3(Dn, Bn), 256, 0, stream>>>(proj, m1, m0, msf, msb);

  // global loss + combine
  final_loss<<<1, 256, 0, stream>>>(msf, msb, cnt, blv, incv, (float*)d_out);
}